// GINEBlock_19765439496859
// MI455X (gfx1250) — compile-verified
//
#include <hip/hip_runtime.h>
#include <hip/hip_bf16.h>

typedef float v2f __attribute__((ext_vector_type(2)));
typedef float v8f __attribute__((ext_vector_type(8)));

#define DIM 128

// ---------------------------------------------------------------------------
// Zero-fill the aggregation workspace (float4 vectorized).
// ---------------------------------------------------------------------------
__global__ __launch_bounds__(256) void gine_zero_kernel(float* __restrict__ p, int n)
{
    int i = (blockIdx.x * 256 + threadIdx.x) * 4;
    if (i + 3 < n) {
        *(float4*)(p + i) = make_float4(0.f, 0.f, 0.f, 0.f);
    } else {
        for (int j = i; j < n; ++j) p[j] = 0.f;
    }
}

// ---------------------------------------------------------------------------
// Edge kernel: msg = relu(x[src] + edge_attr @ We^T + be); atomic scatter to
// aggr[dst]. One wave32 per 32-edge tile (two 16-row A tiles share every
// B fetch -> 16 WMMAs per k-step against 2 A loads + 8 B loads).
// We is staged in LDS as (k,k+1) float2 pairs indexed [k/2][n] so each
// B-operand fetch is a single ds_load_b64.
// ---------------------------------------------------------------------------
__global__ __launch_bounds__(256) void gine_edge_kernel(
    const float* __restrict__ x,
    const int*   __restrict__ ei,        // [2,E] flattened: src row then dst row
    const float* __restrict__ edge_attr, // [E,128]
    const float* __restrict__ We,        // [128,128] row-major [out][in]
    const float* __restrict__ be,        // [128]
    float*       __restrict__ aggr,      // [N,128]
    int E)
{
    __shared__ float2 sWeP[64 * 128];    // 64 KB: sWeP[k2*128+n] = (We[n][2k2], We[n][2k2+1])

    for (int i = threadIdx.x; i < 64 * 128; i += 256) {
        int n  = i >> 6;
        int k2 = i & 63;
        sWeP[k2 * 128 + n] = *(const float2*)(We + n * 128 + 2 * k2);
    }
    __syncthreads();

    const int wave = threadIdx.x >> 5;
    const int lane = threadIdx.x & 31;
    const int row  = lane & 15;
    const int half = lane >> 4;

    int tile = blockIdx.x * 8 + wave;
    int e0   = tile * 32;
    if (e0 >= E) return;

    int myE0 = e0 + row;        if (myE0 >= E) myE0 = E - 1;
    int myE1 = e0 + 16 + row;   if (myE1 >= E) myE1 = E - 1;
    const int mySrc0 = ei[myE0];
    const int myDst0 = ei[E + myE0];
    const int mySrc1 = ei[myE1];
    const int myDst1 = ei[E + myE1];

    const float* attrRow0 = edge_attr + (size_t)myE0 * DIM;
    const float* attrRow1 = edge_attr + (size_t)myE1 * DIM;

    v8f acc0[8] = {};
    v8f acc1[8] = {};

    #pragma unroll 2
    for (int kb = 0; kb < DIM; kb += 4) {
        int ka = kb + half * 2;
        float2 a0 = *(const float2*)(attrRow0 + ka);
        float2 a1 = *(const float2*)(attrRow1 + ka);
        v2f A0; A0[0] = a0.x; A0[1] = a0.y;
        v2f A1; A1[0] = a1.x; A1[1] = a1.y;

        // batch all 8 B pairs first so ds loads can overlap the WMMA burst
        const float2* bRow = sWeP + (ka >> 1) * 128 + row;
        float2 b[8];
        #pragma unroll
        for (int nt = 0; nt < 8; ++nt) b[nt] = bRow[nt * 16];

        #pragma unroll
        for (int nt = 0; nt < 8; ++nt) {
            v2f B; B[0] = b[nt].x; B[1] = b[nt].y;
            acc0[nt] = __builtin_amdgcn_wmma_f32_16x16x4_f32(
                false, A0, false, B, (short)0, acc0[nt], false, false);
            acc1[nt] = __builtin_amdgcn_wmma_f32_16x16x4_f32(
                false, A1, false, B, (short)0, acc1[nt], false, false);
        }
    }

    // Broadcast per-row src/dst indices (lane r holds row r's indices).
    int sR0[8], dR0[8], ok0[8];
    int sR1[8], dR1[8], ok1[8];
    #pragma unroll
    for (int v = 0; v < 8; ++v) {
        int rr = v + half * 8;
        sR0[v] = __shfl(mySrc0, rr, 32);
        dR0[v] = __shfl(myDst0, rr, 32);
        ok0[v] = (e0 + rr) < E;
        sR1[v] = __shfl(mySrc1, rr, 32);
        dR1[v] = __shfl(myDst1, rr, 32);
        ok1[v] = (e0 + 16 + rr) < E;
    }

    // Epilogue: + be + x[src], relu, atomic scatter-add.
    #pragma unroll
    for (int nt = 0; nt < 8; ++nt) {
        int col = nt * 16 + row;
        float bias = be[col];
        #pragma unroll
        for (int v = 0; v < 8; ++v) {
            float m0 = acc0[nt][v] + bias + x[(size_t)sR0[v] * DIM + col];
            m0 = fmaxf(m0, 0.f);
            if (ok0[v]) atomicAdd(aggr + (size_t)dR0[v] * DIM + col, m0);
            float m1 = acc1[nt][v] + bias + x[(size_t)sR1[v] * DIM + col];
            m1 = fmaxf(m1, 0.f);
            if (ok1[v]) atomicAdd(aggr + (size_t)dR1[v] * DIM + col, m1);
        }
    }
}

// ---------------------------------------------------------------------------
// Node kernel: h0=(1+eps)x+aggr; h1=relu(BN(h0@W1^T+b1)); h2=h1@W2^T+b2;
// out=relu(LN(h2)+x). One wave32 per 16-node tile; intermediates live in the
// wave's private 8 KB LDS slice. B-operand (k,k+1) pairs are contiguous in
// the row-major global weights, so W1/W2 are read straight from L2.
// ---------------------------------------------------------------------------
__global__ __launch_bounds__(256) void gine_node_kernel(
    const float* __restrict__ x,
    const float* __restrict__ aggr,
    const float* __restrict__ epsP,
    const float* __restrict__ W1, const float* __restrict__ b1,
    const float* __restrict__ bn_g, const float* __restrict__ bn_b,
    const float* __restrict__ bn_rm, const float* __restrict__ bn_rv,
    const float* __restrict__ W2, const float* __restrict__ b2,
    const float* __restrict__ ln_g, const float* __restrict__ ln_b,
    float* __restrict__ out, int N)
{
    __shared__ float sH[8][16 * DIM];    // 64 KB total, 8 KB per wave

    const int wave = threadIdx.x >> 5;
    const int lane = threadIdx.x & 31;
    const int row  = lane & 15;
    const int half = lane >> 4;
    float* H = sH[wave];

    int tile = blockIdx.x * 8 + wave;
    int n0   = tile * 16;
    if (n0 >= N) return;

    const float eps1 = 1.0f + epsP[0];

    // Phase A: h0 = (1+eps)*x + aggr -> LDS row-major (coalesced float4 loads)
    for (int r = 0; r < 16; ++r) {
        int node = n0 + r;
        if (node >= N) node = N - 1;
        size_t base = (size_t)node * DIM + lane * 4;
        float4 xv = *(const float4*)(x + base);
        float4 av = *(const float4*)(aggr + base);
        float4 h;
        h.x = eps1 * xv.x + av.x;  h.y = eps1 * xv.y + av.y;
        h.z = eps1 * xv.z + av.z;  h.w = eps1 * xv.w + av.w;
        *(float4*)(H + r * DIM + lane * 4) = h;
    }

    v8f acc[8] = {};

    // GEMM1: h0 @ W1^T
    #pragma unroll 4
    for (int kb = 0; kb < DIM; kb += 4) {
        int ka = kb + half * 2;
        float2 a2 = *(const float2*)(H + row * DIM + ka);
        v2f A; A[0] = a2.x; A[1] = a2.y;
        float2 b[8];
        #pragma unroll
        for (int nt = 0; nt < 8; ++nt)
            b[nt] = *(const float2*)(W1 + (size_t)(nt * 16 + row) * DIM + ka);
        #pragma unroll
        for (int nt = 0; nt < 8; ++nt) {
            v2f B; B[0] = b[nt].x; B[1] = b[nt].y;
            acc[nt] = __builtin_amdgcn_wmma_f32_16x16x4_f32(
                false, A, false, B, (short)0, acc[nt], false, false);
        }
    }

    // Epilogue 1: +b1, BatchNorm(eval), ReLU -> back to LDS
    #pragma unroll
    for (int nt = 0; nt < 8; ++nt) {
        int c = nt * 16 + row;
        float a   = bn_g[c] * rsqrtf(bn_rv[c] + 1e-5f);
        float off = (b1[c] - bn_rm[c]) * a + bn_b[c];
        #pragma unroll
        for (int v = 0; v < 8; ++v) {
            int r = v + half * 8;
            H[r * DIM + c] = fmaxf(acc[nt][v] * a + off, 0.f);
        }
        acc[nt] = (v8f){0.f,0.f,0.f,0.f,0.f,0.f,0.f,0.f};
    }

    // GEMM2: h1 @ W2^T
    #pragma unroll 4
    for (int kb = 0; kb < DIM; kb += 4) {
        int ka = kb + half * 2;
        float2 a2 = *(const float2*)(H + row * DIM + ka);
        v2f A; A[0] = a2.x; A[1] = a2.y;
        float2 b[8];
        #pragma unroll
        for (int nt = 0; nt < 8; ++nt)
            b[nt] = *(const float2*)(W2 + (size_t)(nt * 16 + row) * DIM + ka);
        #pragma unroll
        for (int nt = 0; nt < 8; ++nt) {
            v2f B; B[0] = b[nt].x; B[1] = b[nt].y;
            acc[nt] = __builtin_amdgcn_wmma_f32_16x16x4_f32(
                false, A, false, B, (short)0, acc[nt], false, false);
        }
    }

    // Epilogue 2: + b2 -> LDS
    #pragma unroll
    for (int nt = 0; nt < 8; ++nt) {
        int c = nt * 16 + row;
        float bb = b2[c];
        #pragma unroll
        for (int v = 0; v < 8; ++v) {
            int r = v + half * 8;
            H[r * DIM + c] = acc[nt][v] + bb;
        }
    }

    // LayerNorm + residual + relu. Lane (row,half) owns row `row`,
    // columns [half*64, half*64+64); halves combine via shfl_xor 16.
    {
        int node = n0 + row;
        const float4* H4 = (const float4*)(H + row * DIM + half * 64);
        float s1 = 0.f, s2 = 0.f;
        #pragma unroll
        for (int j = 0; j < 16; ++j) {
            float4 t = H4[j];
            s1 += t.x + t.y + t.z + t.w;
            s2 += t.x * t.x + t.y * t.y + t.z * t.z + t.w * t.w;
        }
        s1 += __shfl_xor(s1, 16, 32);
        s2 += __shfl_xor(s2, 16, 32);
        float mu   = s1 * (1.f / 128.f);
        float var  = s2 * (1.f / 128.f) - mu * mu;
        float rstd = rsqrtf(var + 1e-5f);

        if (node < N) {
            #pragma unroll
            for (int j = 0; j < 16; ++j) {
                int cb = half * 64 + j * 4;
                float4 t  = H4[j];
                float4 xv = *(const float4*)(x + (size_t)node * DIM + cb);
                float4 o;
                o.x = fmaxf((t.x - mu) * rstd * ln_g[cb + 0] + ln_b[cb + 0] + xv.x, 0.f);
                o.y = fmaxf((t.y - mu) * rstd * ln_g[cb + 1] + ln_b[cb + 1] + xv.y, 0.f);
                o.z = fmaxf((t.z - mu) * rstd * ln_g[cb + 2] + ln_b[cb + 2] + xv.z, 0.f);
                o.w = fmaxf((t.w - mu) * rstd * ln_g[cb + 3] + ln_b[cb + 3] + xv.w, 0.f);
                *(float4*)(out + (size_t)node * DIM + cb) = o;
            }
        }
    }
}

// ---------------------------------------------------------------------------
extern "C" void kernel_launch(void* const* d_in, const int* in_sizes, int n_in,
                              void* d_out, int out_size, void* d_ws, size_t ws_size,
                              hipStream_t stream)
{
    const float* x         = (const float*)d_in[0];
    const int*   ei        = (const int*)  d_in[1];
    const float* edge_attr = (const float*)d_in[2];
    const float* eps       = (const float*)d_in[3];
    const float* We        = (const float*)d_in[4];
    const float* be        = (const float*)d_in[5];
    const float* W1        = (const float*)d_in[6];
    const float* b1        = (const float*)d_in[7];
    const float* bn_g      = (const float*)d_in[8];
    const float* bn_b      = (const float*)d_in[9];
    const float* bn_rm     = (const float*)d_in[10];
    const float* bn_rv     = (const float*)d_in[11];
    const float* W2        = (const float*)d_in[12];
    const float* b2        = (const float*)d_in[13];
    const float* ln_g      = (const float*)d_in[14];
    const float* ln_b      = (const float*)d_in[15];
    float* out = (float*)d_out;

    const int N = in_sizes[0] / DIM;
    const int E = in_sizes[2] / DIM;

    float* aggr = (float*)d_ws;   // N*DIM floats of scratch

    int total = N * DIM;
    int zblocks = (total / 4 + 255) / 256;
    gine_zero_kernel<<<zblocks, 256, 0, stream>>>(aggr, total);

    int etiles = (E + 31) / 32;                   // 32 edges per wave
    gine_edge_kernel<<<(etiles + 7) / 8, 256, 0, stream>>>(
        x, ei, edge_attr, We, be, aggr, E);

    int ntiles = (N + 15) / 16;
    gine_node_kernel<<<(ntiles + 7) / 8, 256, 0, stream>>>(
        x, aggr, eps, W1, b1, bn_g, bn_b, bn_rm, bn_rv, W2, b2, ln_g, ln_b, out, N);
}